// PCE_51479478010075
// MI455X (gfx1250) — compile-verified
//
#include <hip/hip_runtime.h>

typedef float v2f __attribute__((ext_vector_type(2)));
typedef float v8f __attribute__((ext_vector_type(8)));

#define NPCE        495      // C(12,4) PCE basis terms
#define KPAD        496      // padded K: 124 chunks of 4
#define NCHUNK      124
#define MOUT        256
#define ROWS_PER_WG 64
#define PSI_STRIDE  498      // stride 498 dwords -> bank step 50 (mod 64), conflict-free across 16 lanes

// ---------------------------------------------------------------------------
// Reorder weight[256][495] into per-WMMA B fragments so the GEMM inner loop
// does fully coalesced global_load_b64.
// Fragment f = coltile*124 + chunk: 32 lanes x 2 floats.
//   lane l<16 : B[k+0..k+1][col0+l],  lane 16+l : B[k+2..k+3][col0+l]
// (mirrors the documented 16x4 f32 A-matrix layout)
// ---------------------------------------------------------------------------
__global__ void __launch_bounds__(256) reorder_w(const float* __restrict__ W,
                                                 float* __restrict__ Bf) {
  int t = blockIdx.x * 256 + threadIdx.x;
  const int total = 16 * NCHUNK * 32;
  if (t >= total) return;
  int lane = t & 31;
  int f    = t >> 5;
  int q    = f % NCHUNK;
  int ct   = f / NCHUNK;
  int half = lane >> 4, l15 = lane & 15;
  int col  = ct * 16 + l15;
  int k    = q * 4 + half * 2;
  float b0 = (k     < NPCE) ? W[col * NPCE + k]     : 0.f;
  float b1 = (k + 1 < NPCE) ? W[col * NPCE + k + 1] : 0.f;
  Bf[t * 2 + 0] = b0;
  Bf[t * 2 + 1] = b1;
}

// ---------------------------------------------------------------------------
// Fused kernel: Hermite tables -> psi tile in LDS -> fp32 WMMA GEMM.
// Block = 256 threads = 8 wave32; covers 64 rows x 256 cols of the output.
// Wave w: col tiles [2w, 2w+1], all 4 row tiles -> 8 v8f accumulators.
// ---------------------------------------------------------------------------
template <bool USE_WS>
__global__ void __launch_bounds__(256) pce_gemm(const float* __restrict__ x,
                                                const int*   __restrict__ mi,
                                                const float* __restrict__ W,
                                                const float* __restrict__ Bf,
                                                float*       __restrict__ out) {
  extern __shared__ float smem[];
  float*    psi_lds  = smem;                                   // 64*498 f32
  float*    h_lds    = smem + ROWS_PER_WG * PSI_STRIDE;        // 64*40  f32
  unsigned* pack_lds = (unsigned*)(h_lds + ROWS_PER_WG * 40);  // 495 u32

  const int tid  = threadIdx.x;
  const int row0 = blockIdx.x * ROWS_PER_WG;

  // ---- Stage 0: compact multi-indices. Total degree <= 4 => at most 4
  // non-zero orders; pack each as a byte (dim*5 + order). Unused bytes are 0,
  // which indexes He_0 of dim 0 == 1.0, so a flat 4-way product is exact.
  for (int k = tid; k < NPCE; k += 256) {
    const int* m = mi + k * 8;
    unsigned pk = 0;
    int slot = 0;
    #pragma unroll
    for (int j = 0; j < 8; ++j) {
      int o = m[j];
      if (o > 0) { pk |= (unsigned)(j * 5 + o) << (8 * slot); ++slot; }
    }
    pack_lds[k] = pk;
  }

  // ---- Stage 1: normalized probabilists' Hermite table per (row, dim).
  for (int t = tid; t < ROWS_PER_WG * 8; t += 256) {
    int row = t >> 3, dim = t & 7;
    float xv = x[(size_t)(row0 + row) * 8 + dim];
    float h1 = xv;
    float h2 = xv * h1 - 1.f;
    float h3 = xv * h2 - 2.f * h1;
    float h4 = xv * h3 - 3.f * h2;
    float* hp = h_lds + row * 40 + dim * 5;
    hp[0] = 1.f;
    hp[1] = h1;
    hp[2] = h2 * 0.7071067811865475f;   // 1/sqrt(2!)
    hp[3] = h3 * 0.4082482904638631f;   // 1/sqrt(3!)
    hp[4] = h4 * 0.2041241452319315f;   // 1/sqrt(4!)
  }
  __syncthreads();

  // ---- Stage 2: psi tile [64 x 496] (k = 495 zero-padded).
  for (int t = tid; t < ROWS_PER_WG * KPAD; t += 256) {
    int row = t / KPAD;
    int k   = t - row * KPAD;
    float p = 0.f;
    if (k < NPCE) {
      unsigned pk = pack_lds[k];
      const float* hp = h_lds + row * 40;
      p = hp[pk & 255] * hp[(pk >> 8) & 255] * hp[(pk >> 16) & 255] * hp[pk >> 24];
    }
    psi_lds[row * PSI_STRIDE + k] = p;
  }
  __syncthreads();

  // ---- Stage 3: GEMM via V_WMMA_F32_16X16X4_F32.
  const int lane = tid & 31;
  const int wv   = tid >> 5;        // wave 0..7
  const int half = lane >> 4;       // 0: K=0,1   1: K=2,3
  const int l15  = lane & 15;
  const int ct0  = wv * 2;
  const int col0 = wv * 32;

  v8f acc[4][2] = {};

  int abase[4];
  #pragma unroll
  for (int r = 0; r < 4; ++r)
    abase[r] = (r * 16 + l15) * PSI_STRIDE + half * 2;

  for (int q = 0; q < NCHUNK; ++q) {
    v2f b0, b1;
    if (USE_WS) {
      b0 = *(const v2f*)(Bf + (size_t)(((ct0 + 0) * NCHUNK + q) * 32 + lane) * 2);
      b1 = *(const v2f*)(Bf + (size_t)(((ct0 + 1) * NCHUNK + q) * 32 + lane) * 2);
    } else {
      int k    = q * 4 + half * 2;
      int colA = (ct0 + 0) * 16 + l15;
      int colB = (ct0 + 1) * 16 + l15;
      if (k + 1 < NPCE) {
        b0.x = W[colA * NPCE + k]; b0.y = W[colA * NPCE + k + 1];
        b1.x = W[colB * NPCE + k]; b1.y = W[colB * NPCE + k + 1];
      } else {
        b0.x = (k < NPCE) ? W[colA * NPCE + k] : 0.f; b0.y = 0.f;
        b1.x = (k < NPCE) ? W[colB * NPCE + k] : 0.f; b1.y = 0.f;
      }
    }
    #pragma unroll
    for (int r = 0; r < 4; ++r) {
      v2f a = *(const v2f*)(psi_lds + abase[r] + q * 4);
      acc[r][0] = __builtin_amdgcn_wmma_f32_16x16x4_f32(
          false, a, false, b0, (short)0, acc[r][0], false, false);
      acc[r][1] = __builtin_amdgcn_wmma_f32_16x16x4_f32(
          false, a, false, b1, (short)0, acc[r][1], false, false);
    }
  }

  // ---- Epilogue: C/D layout -> VGPR i: lanes 0-15 M=i, lanes 16-31 M=8+i.
  #pragma unroll
  for (int r = 0; r < 4; ++r) {
    #pragma unroll
    for (int c = 0; c < 2; ++c) {
      #pragma unroll
      for (int i = 0; i < 8; ++i) {
        int row = row0 + r * 16 + half * 8 + i;
        int col = col0 + c * 16 + l15;
        out[(size_t)row * MOUT + col] = acc[r][c][i];
      }
    }
  }
}

// ---------------------------------------------------------------------------
extern "C" void kernel_launch(void* const* d_in, const int* in_sizes, int n_in,
                              void* d_out, int out_size, void* d_ws, size_t ws_size,
                              hipStream_t stream) {
  const float* x  = (const float*)d_in[0];
  const float* W  = (const float*)d_in[1];
  const int*   mi = (const int*)d_in[2];
  float*       out = (float*)d_out;

  const int n_rows = in_sizes[0] / 8;              // 131072
  const int grid   = n_rows / ROWS_PER_WG;         // 2048
  const size_t lds_bytes =
      (size_t)(ROWS_PER_WG * PSI_STRIDE + ROWS_PER_WG * 40 + NPCE) * sizeof(float);
  const size_t bf_bytes = (size_t)16 * NCHUNK * 32 * 2 * sizeof(float);  // ~508 KB

  if (d_ws && ws_size >= bf_bytes) {
    float* Bf = (float*)d_ws;
    reorder_w<<<(16 * NCHUNK * 32 + 255) / 256, 256, 0, stream>>>(W, Bf);
    pce_gemm<true><<<grid, 256, lds_bytes, stream>>>(x, mi, W, Bf, out);
  } else {
    pce_gemm<false><<<grid, 256, lds_bytes, stream>>>(x, mi, W, nullptr, out);
  }
}